// MultiHeadAttention_42855183679992
// MI455X (gfx1250) — compile-verified
//
#include <hip/hip_runtime.h>
#include <stdint.h>

#define B_   64
#define L_   512
#define DIN  256
#define H_   16
#define DK_  32
#define DV_  16
#define HD_  (H_ * DK_)   // 512

typedef __attribute__((ext_vector_type(2))) float v2f;
typedef __attribute__((ext_vector_type(8))) float v8f;

__device__ __forceinline__ v8f wmma_f32(v2f a, v2f b, v8f c) {
  // D = A(16x4,f32) * B(4x16,f32) + C(16x16,f32)
  return __builtin_amdgcn_wmma_f32_16x16x4_f32(false, a, false, b, (short)0, c,
                                               false, false);
}

__device__ __forceinline__ uint32_t lds_addr(const void* p) {
  // generic LDS pointer: addr[31:0] is the wave-relative LDS offset (ISA 10.2)
  return (uint32_t)(uintptr_t)p;
}

__device__ __forceinline__ void async_copy_b128(uint32_t lds, const void* gptr) {
  asm volatile("global_load_async_to_lds_b128 %0, %1, off"
               :
               : "v"(lds), "v"(gptr)
               : "memory");
}

__device__ __forceinline__ void wait_asynccnt0() {
  asm volatile("s_wait_asynccnt 0x0" ::: "memory");
}

// ---------------------------------------------------------------------------
// Kernel 1: Q = v @ Wq + bq ; K = v @ Wk + bk   (blockIdx.z selects Q vs K)
// v: [B*L, 256] row-major, W: [256, 512], out: [B*L, 512]
// Block tile: M=128, N=64, K-step=32, double-buffered.
// A tiles staged with GLOBAL_LOAD_ASYNC_TO_LDS_B128 (TENSOR/ASYNC path),
// W tiles staged transposed so every B fragment is one ds_load_b64.
// ---------------------------------------------------------------------------
__global__ __launch_bounds__(256) void qk_proj_kernel(
    const float* __restrict__ v, const float* __restrict__ Wq,
    const float* __restrict__ bq, const float* __restrict__ Wk,
    const float* __restrict__ bk, float* __restrict__ qbuf,
    float* __restrict__ kbuf) {
  const float* W    = (blockIdx.z == 0) ? Wq : Wk;
  const float* bias = (blockIdx.z == 0) ? bq : bk;
  float*       out  = (blockIdx.z == 0) ? qbuf : kbuf;

  __shared__ __align__(16) float As[2][128][36];   // row pitch 144B (16B mult)
  __shared__ __align__(16) float Bst[2][64][36];   // transposed W tile

  const int tid   = threadIdx.x;
  const int lane  = tid & 31;
  const int wave  = tid >> 5;
  const int lhalf = lane >> 4;
  const int lmod  = lane & 15;

  const int m0g = blockIdx.x * 128;
  const int n0g = blockIdx.y * 64;
  const int m0l = wave * 16;

  // staging index precompute
  const int ar = tid >> 3;            // A: 32 rows per pass, 4 passes
  const int ac = (tid & 7) * 4;
  const int br = tid >> 4;            // B: 16 rows per pass, 2 passes
  const int bc = (tid & 15) * 4;

  v8f acc[4] = {};
  float4 bregs[2];

  // ---- prologue: stage ks = 0 ----
#pragma unroll
  for (int p = 0; p < 4; ++p) {
    const int row = p * 32 + ar;
    async_copy_b128(lds_addr(&As[0][row][ac]),
                    v + (size_t)(m0g + row) * DIN + ac);
  }
#pragma unroll
  for (int p = 0; p < 2; ++p) {
    const int row = p * 16 + br;
    const float4 w = *(const float4*)(W + (size_t)row * HD_ + n0g + bc);
    Bst[0][bc + 0][row] = w.x;
    Bst[0][bc + 1][row] = w.y;
    Bst[0][bc + 2][row] = w.z;
    Bst[0][bc + 3][row] = w.w;
  }
  wait_asynccnt0();
  __syncthreads();

  for (int ks = 0; ks < DIN; ks += 32) {
    const int buf = (ks >> 5) & 1;
    const bool has_next = (ks + 32) < DIN;

    if (has_next) {
      // issue next A tile async + next B tile global loads (overlap compute)
#pragma unroll
      for (int p = 0; p < 4; ++p) {
        const int row = p * 32 + ar;
        async_copy_b128(lds_addr(&As[buf ^ 1][row][ac]),
                        v + (size_t)(m0g + row) * DIN + ks + 32 + ac);
      }
#pragma unroll
      for (int p = 0; p < 2; ++p) {
        const int row = p * 16 + br;
        bregs[p] = *(const float4*)(W + (size_t)(ks + 32 + row) * HD_ + n0g + bc);
      }
    }

    // ---- compute current stage ----
#pragma unroll
    for (int kk = 0; kk < 32; kk += 4) {
      const v2f a = *(const v2f*)&As[buf][m0l + lmod][kk + 2 * lhalf];
#pragma unroll
      for (int t = 0; t < 4; ++t) {
        const v2f bf = *(const v2f*)&Bst[buf][t * 16 + lmod][kk + 2 * lhalf];
        acc[t] = wmma_f32(a, bf, acc[t]);
      }
    }

    if (has_next) {
      // transpose-store next B tile (other buffer; safe while others compute)
#pragma unroll
      for (int p = 0; p < 2; ++p) {
        const int row = p * 16 + br;
        Bst[buf ^ 1][bc + 0][row] = bregs[p].x;
        Bst[buf ^ 1][bc + 1][row] = bregs[p].y;
        Bst[buf ^ 1][bc + 2][row] = bregs[p].z;
        Bst[buf ^ 1][bc + 3][row] = bregs[p].w;
      }
    }
    wait_asynccnt0();
    __syncthreads();
  }

#pragma unroll
  for (int t = 0; t < 4; ++t) {
    const int col = n0g + t * 16 + lmod;
    const float bv = bias[col];
#pragma unroll
    for (int r = 0; r < 8; ++r) {
      const int row = m0g + m0l + 8 * lhalf + r;
      out[(size_t)row * HD_ + col] = acc[t][r] + bv;
    }
  }
}

// ---------------------------------------------------------------------------
// Kernel 2: fused scores + mask + softmax + PV.
// One block per (h, b, 32-query slab). 8 waves.
// ---------------------------------------------------------------------------
__global__ __launch_bounds__(256) void attn_kernel(
    const float* __restrict__ v, const uint8_t* __restrict__ pad,
    const float* __restrict__ qbuf, const float* __restrict__ kbuf,
    float* __restrict__ outp, float* __restrict__ attnp) {
  const int qblk = blockIdx.x;  // 0..15
  const int b    = blockIdx.y;
  const int h    = blockIdx.z;

  __shared__ float Ps[32][520];       // scores/probs, ~65 KB
  __shared__ float Vst[16][520];      // V-head tile transposed, ~33 KB
  __shared__ float Obuf[8][32][16];   // per-wave PV partials, 16 KB
  __shared__ float rowInv[32];

  const int tid   = threadIdx.x;
  const int lane  = tid & 31;
  const int wave  = tid >> 5;
  const int lhalf = lane >> 4;
  const int lmod  = lane & 15;

  const int wrow = wave >> 2;  // 0..1 : 16-row group
  const int wcol = wave & 3;   // 0..3 : 128-key group
  const int q0   = qblk * 32;

  // --- stage V-head tile transposed: Vst[dv][k] = v[b,k,h*16+dv] ---------
  {
#pragma unroll
    for (int p = 0; p < 8; ++p) {
      const int gidx = p * 256 + tid;       // 2048 float4 chunks
      const int row  = gidx >> 2;           // key index 0..511
      const int c4   = (gidx & 3) * 4;      // dv base
      const float4 w =
          *(const float4*)(v + (size_t)(b * L_ + row) * DIN + h * DV_ + c4);
      Vst[c4 + 0][row] = w.x;
      Vst[c4 + 1][row] = w.y;
      Vst[c4 + 2][row] = w.z;
      Vst[c4 + 3][row] = w.w;
    }
  }

  // --- Phase 1: scores S = Q(32x32) @ K^T -> 32 x 512 --------------------
  v2f afrag[8];
  {
    const int qrow = b * L_ + q0 + wrow * 16 + lmod;
    const float* qp = qbuf + (size_t)qrow * HD_ + h * DK_;
#pragma unroll
    for (int kk = 0; kk < 8; ++kk)
      afrag[kk] = *(const v2f*)(qp + kk * 4 + 2 * lhalf);
  }
  const float scale = 0.17677669529663687f;  // 1/sqrt(DK)
#pragma unroll
  for (int t = 0; t < 8; ++t) {
    const int col = wcol * 128 + t * 16 + lmod;  // key index (per lane)
    const float* kp = kbuf + (size_t)(b * L_ + col) * HD_ + h * DK_;
    v8f acc = {};
#pragma unroll
    for (int kk = 0; kk < 8; ++kk) {
      const v2f bf = *(const v2f*)(kp + kk * 4 + 2 * lhalf);  // K^T frag
      acc = wmma_f32(afrag[kk], bf, acc);
    }
    const bool masked = pad[b * L_ + col] != 0;
#pragma unroll
    for (int r = 0; r < 8; ++r) {
      const int rowl = wrow * 16 + 8 * lhalf + r;
      Ps[rowl][col] = masked ? -1000.0f : acc[r] * scale;
    }
  }
  __syncthreads();   // covers Ps and Vst

  // --- Phase 2: softmax over 512 keys per row ----------------------------
  {
    const int row = tid >> 3;   // 8 threads per row
    const int sub = tid & 7;    // 64 cols each
    float* prow = &Ps[row][0];
    float m = -3.4e38f;
    for (int j = 0; j < 64; ++j) m = fmaxf(m, prow[sub * 64 + j]);
#pragma unroll
    for (int o = 4; o >= 1; o >>= 1) m = fmaxf(m, __shfl_xor(m, o, 32));
    float ssum = 0.f;
    for (int j = 0; j < 64; ++j) {
      const float e = __expf(prow[sub * 64 + j] - m);
      prow[sub * 64 + j] = e;
      ssum += e;
    }
#pragma unroll
    for (int o = 4; o >= 1; o >>= 1) ssum += __shfl_xor(ssum, o, 32);
    if (sub == 0) rowInv[row] = 1.0f / ssum;
  }
  __syncthreads();
  {  // normalize + single coalesced attn store
    float* attnbase = attnp + (((size_t)h * B_ + b) * L_ + q0) * L_;
    for (int idx = tid; idx < 32 * 512; idx += 256) {
      const int row = idx >> 9;
      const int col = idx & 511;
      const float p = Ps[row][col] * rowInv[row];
      Ps[row][col] = p;
      attnbase[(size_t)row * L_ + col] = p;
    }
  }
  __syncthreads();

  // --- Phase 3: O = P @ Vh (split K=512 over 8 waves), all from LDS ------
  {
    const int k0 = wave * 64;
#pragma unroll
    for (int mt = 0; mt < 2; ++mt) {
      v8f acc = {};
#pragma unroll
      for (int kk = 0; kk < 64; kk += 4) {
        const v2f a  = *(const v2f*)&Ps[mt * 16 + lmod][k0 + kk + 2 * lhalf];
        const v2f bf = *(const v2f*)&Vst[lmod][k0 + kk + 2 * lhalf];
        acc = wmma_f32(a, bf, acc);
      }
#pragma unroll
      for (int r = 0; r < 8; ++r)
        Obuf[wave][mt * 16 + 8 * lhalf + r][lmod] = acc[r];
    }
  }
  __syncthreads();
  {  // cross-wave reduce + store output [H,B,L,DV]
    float* obase = outp + (((size_t)h * B_ + b) * L_ + q0) * DV_;
    for (int idx = tid; idx < 32 * 16; idx += 256) {
      const int row = idx >> 4;
      const int dv  = idx & 15;
      float s = 0.f;
#pragma unroll
      for (int w = 0; w < 8; ++w) s += Obuf[w][row][dv];
      obase[(size_t)row * DV_ + dv] = s;
    }
  }
}

// ---------------------------------------------------------------------------
extern "C" void kernel_launch(void* const* d_in, const int* in_sizes, int n_in,
                              void* d_out, int out_size, void* d_ws,
                              size_t ws_size, hipStream_t stream) {
  const float*   v   = (const float*)d_in[0];
  const uint8_t* pad = (const uint8_t*)d_in[1];
  const float*   Wq  = (const float*)d_in[2];
  const float*   bq  = (const float*)d_in[3];
  const float*   Wk  = (const float*)d_in[4];
  const float*   bk  = (const float*)d_in[5];

  float* out  = (float*)d_out;                        // [H,B,L,DV]
  float* attn = out + (size_t)H_ * B_ * L_ * DV_;     // [H,B,L,L]

  float* qbuf = (float*)d_ws;                         // [B*L, 512] = 64 MB
  float* kbuf = qbuf + (size_t)B_ * L_ * HD_;         // [B*L, 512] = 64 MB

  qk_proj_kernel<<<dim3(B_ * L_ / 128, HD_ / 64, 2), 256, 0, stream>>>(
      v, Wq, bq, Wk, bk, qbuf, kbuf);
  attn_kernel<<<dim3(L_ / 32, B_, H_), 256, 0, stream>>>(v, pad, qbuf, kbuf,
                                                         out, attn);
}